// SAGE_12927851561477
// MI455X (gfx1250) — compile-verified
//
#include <hip/hip_runtime.h>
#include <math.h>

typedef __attribute__((ext_vector_type(2))) float v2f;
typedef __attribute__((ext_vector_type(8))) float v8f;
typedef __attribute__((ext_vector_type(4))) int   v4i;

// pointer-to-v4i in global (AS1) / LDS (AS3) address spaces for the async DMA builtin
typedef __attribute__((address_space(1))) v4i* global_v4i_p;
typedef __attribute__((address_space(3))) v4i* lds_v4i_p;

#define D 256

#if defined(__has_builtin)
#if __has_builtin(__builtin_amdgcn_global_load_async_to_lds_b128) && \
    __has_builtin(__builtin_amdgcn_s_wait_asynccnt)
#define GEMM_ASYNC 1
#endif
#endif

// ---------------------------------------------------------------------------
// h = log(x + 1), vectorized float4
// ---------------------------------------------------------------------------
__global__ void log1p_kernel(const float4* __restrict__ x, float4* __restrict__ h, int n4) {
    int i = blockIdx.x * blockDim.x + threadIdx.x;
    if (i < n4) {
        float4 v = x[i];
        float4 r;
        r.x = logf(v.x + 1.0f);
        r.y = logf(v.y + 1.0f);
        r.z = logf(v.z + 1.0f);
        r.w = logf(v.w + 1.0f);
        h[i] = r;
    }
}

// ---------------------------------------------------------------------------
// zero fill (float4)
// ---------------------------------------------------------------------------
__global__ void zero_kernel(float4* __restrict__ p, int n4) {
    int i = blockIdx.x * blockDim.x + threadIdx.x;
    if (i < n4) p[i] = make_float4(0.f, 0.f, 0.f, 0.f);
}

// ---------------------------------------------------------------------------
// Fold BN into fc epilogue:  out = acc*a + c
//   a = gamma * rsqrt(var+eps);  c = fc_b*a + beta - mean*a
// blockIdx.x = layer
// ---------------------------------------------------------------------------
__global__ void bnprep_kernel(const float* __restrict__ gamma, const float* __restrict__ beta,
                              const float* __restrict__ mean,  const float* __restrict__ var,
                              const float* __restrict__ fcb,
                              float* __restrict__ a, float* __restrict__ c) {
    int l = blockIdx.x;
    int j = threadIdx.x;
    int idx = l * D + j;
    float s = gamma[idx] * rsqrtf(var[idx] + 1e-5f);
    a[idx] = s;
    c[idx] = fcb[idx] * s + beta[idx] - mean[idx] * s;
}

// ---------------------------------------------------------------------------
// GEMM: out = act( (A @ W^T) * scale + shift + addsrc )
//   A: M x 256 row-major, W: 256 x 256 row-major (row j = weights of out col j)
// Block: 128 threads = 4 waves; each wave computes a 16 x 256 strip using
// V_WMMA_F32_16X16X4_F32 (16 acc tiles in registers, A-frag reused 16x).
// Staging: double-buffered LDS; async DMA (GLOBAL_LOAD_ASYNC_TO_LDS_B128,
// ASYNCcnt-tracked) when the toolchain exposes it, else sync loads+stores.
// ---------------------------------------------------------------------------
__global__ __launch_bounds__(128)
void gemm256_kernel(const float* __restrict__ A, const float* __restrict__ W,
                    const float* __restrict__ scale, const float* __restrict__ shift,
                    const float* addsrc, float* out,
                    int M, int do_relu)
{
    constexpr int KC  = 16;      // k-chunk staged in LDS
    constexpr int LDA = KC + 2;  // pad: stride 18 dwords -> conflict-free frag reads
    constexpr int NC  = D / KC;  // 16 chunks
    __shared__ float sA[2][64 * LDA];   //  64 rows x 16 k, double buffered
    __shared__ float sB[2][D  * LDA];   // 256 cols x 16 k (B[k][n] = W[n][k])

    const int tid   = threadIdx.x;
    const int wave  = tid >> 5;
    const int lane  = tid & 31;
    const int lhalf = lane >> 4;     // 0 or 1
    const int lmod  = lane & 15;
    const int rowB  = blockIdx.x * 64;
    const int row0  = rowB + wave * 16;

    v8f acc[16] = {};

    // ---- one k-chunk of WMMA work from a staged buffer ----
    auto compute = [&](const float* cA, const float* cB) {
        #pragma unroll
        for (int kk = 0; kk < KC; kk += 4) {
            // A fragment: lane l holds A[row0 + l%16, k + 2*(l>>4) + {0,1}]
            const float* ap = &cA[(wave * 16 + lmod) * LDA + kk + 2 * lhalf];
            v2f a;
            a.x = ap[0];
            a.y = ap[1];
            #pragma unroll
            for (int nt = 0; nt < 16; ++nt) {
                // B fragment = W[nt*16 + l%16, k + 2*(l>>4) + {0,1}]
                const float* bp = &cB[(nt * 16 + lmod) * LDA + kk + 2 * lhalf];
                v2f b;
                b.x = bp[0];
                b.y = bp[1];
                acc[nt] = __builtin_amdgcn_wmma_f32_16x16x4_f32(
                    false, a, false, b, (short)0, acc[nt], false, false);
            }
        }
    };

#if GEMM_ASYNC
    // ---- async DMA staging: 10 x b128 per thread per chunk (2 for A, 8 for W)
    auto stage_async = [&](int k0, float* dA, float* dB) {
        #pragma unroll
        for (int i = 0; i < 2; ++i) {               // A: 64 rows x 4 groups of 4 floats
            int g = tid + i * 128;
            int r = g >> 2;
            int c = (g & 3) * 4;
            int gr = rowB + r;
            gr = gr < M ? gr : (M - 1);             // clamp tail (stores guarded)
            const float* src = A + (size_t)gr * D + k0 + c;
            float* dst = dA + r * LDA + c;
            __builtin_amdgcn_global_load_async_to_lds_b128(
                (global_v4i_p)src, (lds_v4i_p)dst, 0, 0);
        }
        #pragma unroll
        for (int i = 0; i < 8; ++i) {               // W: 256 rows x 4 groups
            int g = tid + i * 128;
            int r = g >> 2;
            int c = (g & 3) * 4;
            const float* src = W + (size_t)r * D + k0 + c;
            float* dst = dB + r * LDA + c;
            __builtin_amdgcn_global_load_async_to_lds_b128(
                (global_v4i_p)src, (lds_v4i_p)dst, 0, 0);
        }
    };

    stage_async(0, sA[0], sB[0]);
    for (int c = 0; c < NC; ++c) {
        int cur = c & 1;
        if (c + 1 < NC) {
            stage_async((c + 1) * KC, sA[cur ^ 1], sB[cur ^ 1]);
            __builtin_amdgcn_s_wait_asynccnt(10);   // chunk c done; c+1 in flight
        } else {
            __builtin_amdgcn_s_wait_asynccnt(0);
        }
        __syncthreads();                            // all waves' DMA visible
        compute(sA[cur], sB[cur]);
        __syncthreads();                            // done reading before reuse
    }
#else
    // ---- synchronous staging fallback ----
    for (int c = 0; c < NC; ++c) {
        int k0 = c * KC;
        {
            int r  = tid >> 1;
            int cc = (tid & 1) * 8;
            int gr = rowB + r;
            gr = gr < M ? gr : (M - 1);
            const float4* src = reinterpret_cast<const float4*>(A + (size_t)gr * D + k0 + cc);
            float4 p0 = src[0];
            float4 p1 = src[1];
            float* d = &sA[0][r * LDA + cc];
            d[0] = p0.x; d[1] = p0.y; d[2] = p0.z; d[3] = p0.w;
            d[4] = p1.x; d[5] = p1.y; d[6] = p1.z; d[7] = p1.w;
        }
        #pragma unroll
        for (int it = 0; it < 4; ++it) {
            int idx = it * 128 + tid;
            int r   = idx >> 1;
            int cc  = (idx & 1) * 8;
            const float4* src = reinterpret_cast<const float4*>(W + (size_t)r * D + k0 + cc);
            float4 p0 = src[0];
            float4 p1 = src[1];
            float* d = &sB[0][r * LDA + cc];
            d[0] = p0.x; d[1] = p0.y; d[2] = p0.z; d[3] = p0.w;
            d[4] = p1.x; d[5] = p1.y; d[6] = p1.z; d[7] = p1.w;
        }
        __syncthreads();
        compute(sA[0], sB[0]);
        __syncthreads();
    }
#endif

    // ---- epilogue: C tile: lane<16 -> M=j, lane>=16 -> M=j+8; N = lane%16 ----
    #pragma unroll
    for (int nt = 0; nt < 16; ++nt) {
        int col  = nt * 16 + lmod;
        float sc = scale ? scale[col] : 1.0f;
        float sh = shift ? shift[col] : 0.0f;
        #pragma unroll
        for (int j = 0; j < 8; ++j) {
            int r = row0 + j + lhalf * 8;
            if (r < M) {
                size_t idx = (size_t)r * D + col;
                float v = acc[nt][j] * sc + sh;
                if (addsrc) v += addsrc[idx];
                if (do_relu) v = fmaxf(v, 0.0f);
                out[idx] = v;
            }
        }
    }
}

// ---------------------------------------------------------------------------
// Edge scatter-max: hneigh[dst] = max(hneigh[dst], m[src]) elementwise.
// m is post-ReLU (>=0) so float order == int order on raw bits, and the
// reference's where(deg>0, segmax, 0) is exactly "init 0 + atomicMax(int)".
// One wave per edge; 8 coalesced dwords per lane.
// ---------------------------------------------------------------------------
__global__ __launch_bounds__(256)
void scatter_max_kernel(const float* __restrict__ m, const int* __restrict__ esrc,
                        const int* __restrict__ edst, int* __restrict__ hneigh_i, int E)
{
    int e = blockIdx.x * 8 + (threadIdx.x >> 5);
    if (e >= E) return;
    int lane = threadIdx.x & 31;
    const float* src = m + (size_t)esrc[e] * D;
    int* dst = hneigh_i + (size_t)edst[e] * D;
    #pragma unroll
    for (int i = 0; i < 8; ++i) {
        float v = src[lane + i * 32];
        atomicMax(&dst[lane + i * 32], __float_as_int(v));
    }
}

// ---------------------------------------------------------------------------
// Row-wise L2 normalize: h[r] /= max(||h[r]||, 1e-12). One wave per row.
// ---------------------------------------------------------------------------
__global__ __launch_bounds__(256)
void rownorm_kernel(float* __restrict__ h, int M)
{
    int row = blockIdx.x * 8 + (threadIdx.x >> 5);
    if (row >= M) return;
    int lane = threadIdx.x & 31;
    float4* p = reinterpret_cast<float4*>(h + (size_t)row * D);
    float4 a = p[lane * 2];
    float4 b = p[lane * 2 + 1];
    float ss = a.x * a.x + a.y * a.y + a.z * a.z + a.w * a.w
             + b.x * b.x + b.y * b.y + b.z * b.z + b.w * b.w;
    #pragma unroll
    for (int msk = 16; msk; msk >>= 1) ss += __shfl_xor(ss, msk, 32);
    float inv = 1.0f / fmaxf(sqrtf(ss), 1e-12f);
    a.x *= inv; a.y *= inv; a.z *= inv; a.w *= inv;
    b.x *= inv; b.y *= inv; b.z *= inv; b.w *= inv;
    p[lane * 2]     = a;
    p[lane * 2 + 1] = b;
}

// ---------------------------------------------------------------------------
extern "C" void kernel_launch(void* const* d_in, const int* in_sizes, int n_in,
                              void* d_out, int out_size, void* d_ws, size_t ws_size,
                              hipStream_t stream)
{
    const float* x       = (const float*)d_in[0];
    const float* pool_W  = (const float*)d_in[1];
    const float* pool_b  = (const float*)d_in[2];
    const float* neigh_W = (const float*)d_in[3];
    const float* self_W  = (const float*)d_in[4];
    const float* sage_b  = (const float*)d_in[5];
    const float* fc_W    = (const float*)d_in[6];
    const float* fc_b    = (const float*)d_in[7];
    const float* bn_g    = (const float*)d_in[8];
    const float* bn_b    = (const float*)d_in[9];
    const float* bn_m    = (const float*)d_in[10];
    const float* bn_v    = (const float*)d_in[11];
    const int*   e_src0  = (const int*)d_in[12];
    const int*   e_dst0  = (const int*)d_in[13];
    const int*   e_src1  = (const int*)d_in[14];
    const int*   e_dst1  = (const int*)d_in[15];
    float*       outp    = (float*)d_out;

    const int N_SRC0 = 150000, N_DST0 = 50000, N_DST1 = 12800;
    const int E0 = 500000, E1 = 128000;
    const int WMAT = D * D; // 65536

    // workspace layout (floats)
    float* ws     = (float*)d_ws;
    float* h0     = ws;                               // 150000*256
    float* mbuf   = h0     + (size_t)N_SRC0 * D;      // 150000*256
    float* hneigh = mbuf   + (size_t)N_SRC0 * D;      //  50000*256
    float* tmp    = hneigh + (size_t)N_DST0 * D;      //  50000*256
    float* bnA    = tmp    + (size_t)N_DST0 * D;      //  2*256
    float* bnC    = bnA    + 2 * D;                   //  2*256

    // h0 = log(x + 1)
    {
        int n4 = N_SRC0 * D / 4;
        log1p_kernel<<<(n4 + 255) / 256, 256, 0, stream>>>((const float4*)x, (float4*)h0, n4);
    }
    // fold BN for both layers
    bnprep_kernel<<<2, D, 0, stream>>>(bn_g, bn_b, bn_m, bn_v, fc_b, bnA, bnC);

    // ======================= layer 0 =======================
    gemm256_kernel<<<(N_SRC0 + 63) / 64, 128, 0, stream>>>(
        h0, pool_W, nullptr, pool_b, nullptr, mbuf, N_SRC0, 1);
    {
        int n4 = N_DST0 * D / 4;
        zero_kernel<<<(n4 + 255) / 256, 256, 0, stream>>>((float4*)hneigh, n4);
    }
    scatter_max_kernel<<<E0 / 8, 256, 0, stream>>>(mbuf, e_src0, e_dst0, (int*)hneigh, E0);
    gemm256_kernel<<<(N_DST0 + 63) / 64, 128, 0, stream>>>(
        h0, self_W, nullptr, nullptr, nullptr, tmp, N_DST0, 0);
    gemm256_kernel<<<(N_DST0 + 63) / 64, 128, 0, stream>>>(
        hneigh, neigh_W, nullptr, sage_b, tmp, tmp, N_DST0, 1);
    rownorm_kernel<<<(N_DST0 + 7) / 8, 256, 0, stream>>>(tmp, N_DST0);
    gemm256_kernel<<<(N_DST0 + 63) / 64, 128, 0, stream>>>(
        tmp, fc_W, bnA, bnC, nullptr, h0, N_DST0, 1);

    // ======================= layer 1 =======================
    gemm256_kernel<<<(N_DST0 + 63) / 64, 128, 0, stream>>>(
        h0, pool_W + WMAT, nullptr, pool_b + D, nullptr, mbuf, N_DST0, 1);
    {
        int n4 = N_DST1 * D / 4;
        zero_kernel<<<(n4 + 255) / 256, 256, 0, stream>>>((float4*)hneigh, n4);
    }
    scatter_max_kernel<<<E1 / 8, 256, 0, stream>>>(mbuf, e_src1, e_dst1, (int*)hneigh, E1);
    gemm256_kernel<<<(N_DST1 + 63) / 64, 128, 0, stream>>>(
        h0, self_W + WMAT, nullptr, nullptr, nullptr, tmp, N_DST1, 0);
    gemm256_kernel<<<(N_DST1 + 63) / 64, 128, 0, stream>>>(
        hneigh, neigh_W + WMAT, nullptr, sage_b + D, tmp, tmp, N_DST1, 1);
    rownorm_kernel<<<(N_DST1 + 7) / 8, 256, 0, stream>>>(tmp, N_DST1);
    gemm256_kernel<<<(N_DST1 + 63) / 64, 128, 0, stream>>>(
        tmp, fc_W + WMAT, bnA + D, bnC + D, nullptr, outp, N_DST1, 0);
}